// Rendering_86363202388361
// MI455X (gfx1250) — compile-verified
//
#include <hip/hip_runtime.h>
#include <stdint.h>

// ---------------------------------------------------------------------------
// NeRF-style volume rendering composite for MI455X (gfx1250, wave32).
//
// Roofline: 674 MB streamed / 23.3 TB/s  ->  ~29 us floor. Memory bound.
// Strategy: one wave per ray, double-buffered ASYNC global->LDS staging
// (CDNA5 ASYNCcnt path) with non-temporal hints (one-pass stream vs 192MB L2),
// swizzled LDS layout, wave32 shuffle scan/reduce.
// ---------------------------------------------------------------------------

#define NERF_S            128
#define FAR_DIST          1e10f
#define EPS_T             1e-10f

#define WAVES_PER_BLOCK   8
#define BLOCK_THREADS     (WAVES_PER_BLOCK * 32)
#define NUM_BLOCKS        2048

#define RAW_BYTES_PER_RAY (NERF_S * 16)                    // 2048
#define Z_BYTES_PER_RAY   (NERF_S * 4)                     // 512
#define BUF_BYTES         (RAW_BYTES_PER_RAY + Z_BYTES_PER_RAY)  // 2560
#define WAVE_LDS          (2 * BUF_BYTES)                  // 5120 (double buffer)
#define BLOCK_LDS         (WAVES_PER_BLOCK * WAVE_LDS)     // 40960

// ---- CDNA5 async global->LDS copy (per-lane 16B), tracked by ASYNCcnt -----
// Data is streamed exactly once (674MB > 192MB L2): mark non-temporal.
__device__ __forceinline__ void async_copy_b128(unsigned lds_byte,
                                                const void* sbase,
                                                unsigned voff) {
  // GVS mode: mem_addr = SGPR64 + VGPR32 offset; VDST = LDS byte address.
  asm volatile("global_load_async_to_lds_b128 %0, %1, %2 th:TH_LOAD_NT"
               :
               : "v"(lds_byte), "v"(voff), "s"(sbase)
               : "memory");
}

__device__ __forceinline__ void wait_async_le5() {
  asm volatile("s_wait_asynccnt 0x5" ::: "memory");
}
__device__ __forceinline__ void wait_async_0() {
  asm volatile("s_wait_asynccnt 0x0" ::: "memory");
}
__device__ __forceinline__ void wait_ds_0() {
  asm volatile("s_wait_dscnt 0x0" ::: "memory");
}

// 16B-chunk XOR swizzle: makes both async writes and per-lane b128 LDS reads
// hit distinct bank groups (lane L reads chunks 4L..4L+3).
__device__ __forceinline__ unsigned swz(unsigned g) {
  return g ^ ((g >> 4) & 3u);
}

__device__ __forceinline__ float sigmoid_fast(float x) {
  return __builtin_amdgcn_rcpf(1.0f + __expf(-x));
}

// Issue the 5 async b128 copies (4 x raw, 1 x z) for one ray into one buffer.
__device__ __forceinline__ void prefetch_ray(const float* __restrict__ raw,
                                             const float* __restrict__ z,
                                             int ray, unsigned lds_raw,
                                             unsigned lds_z, int lane) {
  const void* graw = (const char*)raw + (size_t)ray * RAW_BYTES_PER_RAY;
  const void* gz   = (const char*)z   + (size_t)ray * Z_BYTES_PER_RAY;
#pragma unroll
  for (int i = 0; i < 4; ++i) {
    unsigned g = (unsigned)(i * 32 + lane);          // global chunk: contiguous 512B/instr
    async_copy_b128(lds_raw + swz(g) * 16u, graw, g * 16u);
  }
  async_copy_b128(lds_z + (unsigned)lane * 16u, gz, (unsigned)lane * 16u);
}

__global__ __launch_bounds__(BLOCK_THREADS)
void nerf_render_kernel(const float* __restrict__ raw,
                        const float* __restrict__ z_vals,
                        float* __restrict__ out, int num_rays) {
  __shared__ __align__(16) unsigned char smem[BLOCK_LDS];

  const int lane  = threadIdx.x & 31;
  const int wslot = threadIdx.x >> 5;
  const int wave  = blockIdx.x * WAVES_PER_BLOCK + wslot;
  const int nwaves = gridDim.x * WAVES_PER_BLOCK;

  const unsigned base = (unsigned)wslot * WAVE_LDS;
  const unsigned lds0 = (unsigned)(uintptr_t)(&smem[0]);  // LDS byte offset of smem

  const int r0 = wave;
  if (r0 < num_rays) {
    prefetch_ray(raw, z_vals, r0, lds0 + base, lds0 + base + 4096u, lane);
  }

  int cur = 0;
  for (int r = r0; r < num_rays; r += nwaves) {
    const int nr = r + nwaves;
    const unsigned roff_cur = base + (cur ? 2048u : 0u);
    const unsigned zoff_cur = base + 4096u + (cur ? 512u : 0u);

    if (nr < num_rays) {
      // Buffer we are about to overwrite was fully consumed (ds reads) last iter.
      wait_ds_0();
      const unsigned roff_nxt = base + (cur ? 0u : 2048u);
      const unsigned zoff_nxt = base + 4096u + (cur ? 0u : 512u);
      prefetch_ray(raw, z_vals, nr, lds0 + roff_nxt, lds0 + zoff_nxt, lane);
      wait_async_le5();   // 10 outstanding; in-order retire -> current buf done
    } else {
      wait_async_0();
    }

    // ---------------- compute ray r from buffer `cur` ----------------
    const unsigned char* rbuf = smem + roff_cur;
    const unsigned char* zbuf = smem + zoff_cur;

    const float4 z4 = *(const float4*)(zbuf + (unsigned)lane * 16u);
    const float nz = __shfl_down(z4.x, 1, 32);       // z[4L+4] from lane L+1
    float dj[4];
    dj[0] = z4.y - z4.x;
    dj[1] = z4.z - z4.y;
    dj[2] = z4.w - z4.z;
    dj[3] = (lane == 31) ? FAR_DIST : (nz - z4.w);

    float4 rv[4];
#pragma unroll
    for (int j = 0; j < 4; ++j) {
      const unsigned g = (unsigned)(lane * 4 + j);   // sample index == chunk index
      rv[j] = *(const float4*)(rbuf + swz(g) * 16u);
    }

    float a[4], t[4];
#pragma unroll
    for (int j = 0; j < 4; ++j) {
      const float sigma = rv[j].w > 0.0f ? rv[j].w : 0.0f;  // relu
      const float e = __expf(-sigma * dj[j]);
      a[j] = 1.0f - e;            // alpha
      t[j] = e + EPS_T;           // (1 - alpha) + eps
    }

    // In-lane exclusive prefixes and lane product.
    const float q1 = t[0];
    const float q2 = t[0] * t[1];
    const float q3 = q2 * t[2];
    const float P  = q3 * t[3];

    // Wave32 multiplicative Hillis-Steele inclusive scan of P.
    float v = P;
#pragma unroll
    for (int d = 1; d < 32; d <<= 1) {
      const float u = __shfl_up(v, (unsigned)d, 32);
      if (lane >= d) v *= u;
    }
    float E = __shfl_up(v, 1u, 32);                  // exclusive scan
    if (lane == 0) E = 1.0f;

    const float w0 = a[0] * E;
    const float w1 = a[1] * (E * q1);
    const float w2 = a[2] * (E * q2);
    const float w3 = a[3] * (E * q3);

    float sr = w0 * sigmoid_fast(rv[0].x) + w1 * sigmoid_fast(rv[1].x) +
               w2 * sigmoid_fast(rv[2].x) + w3 * sigmoid_fast(rv[3].x);
    float sg = w0 * sigmoid_fast(rv[0].y) + w1 * sigmoid_fast(rv[1].y) +
               w2 * sigmoid_fast(rv[2].y) + w3 * sigmoid_fast(rv[3].y);
    float sb = w0 * sigmoid_fast(rv[0].z) + w1 * sigmoid_fast(rv[1].z) +
               w2 * sigmoid_fast(rv[2].z) + w3 * sigmoid_fast(rv[3].z);
    float sa = w0 + w1 + w2 + w3;

#pragma unroll
    for (int d = 16; d >= 1; d >>= 1) {
      sr += __shfl_xor(sr, d, 32);
      sg += __shfl_xor(sg, d, 32);
      sb += __shfl_xor(sb, d, 32);
      sa += __shfl_xor(sa, d, 32);
    }

    // After the butterfly every lane holds the totals: lanes 0..2 each store
    // one channel -> a single coalesced global_store_b32 per ray.
    const float bg = 1.0f - sa;
    const float chan = (lane == 0) ? (sr + bg) : ((lane == 1) ? (sg + bg) : (sb + bg));
    if (lane < 3) {
      out[(size_t)r * 3 + lane] = chan;
    }

    cur ^= 1;
  }
}

extern "C" void kernel_launch(void* const* d_in, const int* in_sizes, int n_in,
                              void* d_out, int out_size, void* d_ws, size_t ws_size,
                              hipStream_t stream) {
  (void)n_in; (void)out_size; (void)d_ws; (void)ws_size;
  const float* raw    = (const float*)d_in[0];   // [R, 128, 4] f32
  const float* z_vals = (const float*)d_in[1];   // [R, 128]    f32
  float* out          = (float*)d_out;           // [R, 3]      f32
  const int num_rays  = in_sizes[1] / NERF_S;

  nerf_render_kernel<<<dim3(NUM_BLOCKS), dim3(BLOCK_THREADS), 0, stream>>>(
      raw, z_vals, out, num_rays);
}